// MOELayer_68453188764124
// MI455X (gfx1250) — compile-verified
//
#include <hip/hip_runtime.h>
#include <hip/hip_bf16.h>
#include <math.h>

#define DIM      128
#define NEXP     4
#define NGRAPH   256
#define EPSV     1e-5f

typedef float v2f __attribute__((ext_vector_type(2)));
typedef float v8f __attribute__((ext_vector_type(8)));

// ---------------------------------------------------------------- utilities
__device__ __forceinline__ void atomic_max_f32(float* addr, float v) {
    // monotone int trick: works for mixed signs with -inf init
    if (v >= 0.0f) atomicMax((int*)addr, __float_as_int(v));
    else           atomicMin((unsigned int*)addr, (unsigned int)__float_as_int(v));
}

__global__ void k_fill(float* p, long long n, float v) {
    long long i = (long long)blockIdx.x * blockDim.x + threadIdx.x;
    if (i < n) p[i] = v;
}

// -------------------------------------------- weight repack to B-fragment
// Input: W row-major [K, Ncols]. Output pairs Wp[p], where
//   p = ((j*(K/4) + s2)*2 + half)*16 + c
//   pair = ( W[4*s2+2*half][j*16+c], W[4*s2+2*half+1][j*16+c] )
// so the hot-loop B-fragment load is ONE coalesced b64 per lane.
__global__ void k_repack(const float* __restrict__ W, int K, int Ncols,
                         v2f* __restrict__ Wp) {
    int p = blockIdx.x * blockDim.x + threadIdx.x;
    int total = (K / 2) * Ncols;       // number of pairs
    if (p >= total) return;
    int c    = p & 15;
    int p2   = p >> 4;
    int half = p2 & 1;
    int p3   = p2 >> 1;
    int kq   = K >> 2;
    int s2   = p3 % kq;
    int j    = p3 / kq;
    int krow = 4 * s2 + 2 * half;
    int col  = j * 16 + c;
    v2f out;
    out.x = W[(long long)krow * Ncols + col];
    out.y = W[(long long)(krow + 1) * Ncols + col];
    Wp[p] = out;
}

// ---------------------------------------------------------------- degrees
__global__ void k_deg(const int* __restrict__ dst, int E, float* __restrict__ deg) {
    int e = blockIdx.x * blockDim.x + threadIdx.x;
    if (e < E) atomicAdd(&deg[dst[e]], 1.0f);
}

// ------------------------------------------------- per-graph pooling stats
__global__ void k_node_stats(const float* __restrict__ x, const int* __restrict__ batch,
                             const float* __restrict__ deg, int N,
                             float* gsum, float* gmax, float* gcnt,
                             float* gdsum, float* gdmax) {
    long long i = (long long)blockIdx.x * blockDim.x + threadIdx.x;
    if (i >= (long long)N * DIM) return;
    int n = (int)(i / DIM), d = (int)(i % DIM);
    int b = batch[n];
    float v = x[i];
    atomicAdd(&gsum[b * DIM + d], v);
    atomic_max_f32(&gmax[b * DIM + d], v);
    if (d == 0) {
        atomicAdd(&gcnt[b], 1.0f);
        atomicAdd(&gdsum[b], deg[n]);
        atomic_max_f32(&gdmax[b], deg[n]);
    }
}

// ----------------------------------------------------- router MLP + softmax
__global__ void k_router(const float* gsum, const float* gmax, const float* gcnt,
                         const float* gdsum, const float* gdmax,
                         const float* __restrict__ temp,
                         const float* __restrict__ rw1, const float* __restrict__ rb1,
                         const float* __restrict__ rw2, const float* __restrict__ rb2,
                         const float* __restrict__ rw3, const float* __restrict__ rb3,
                         float* gates) {
    int b = blockIdx.x * blockDim.x + threadIdx.x;
    if (b >= NGRAPH) return;
    float cnt = fmaxf(gcnt[b], 1.0f);
    float stats[4] = { cnt, gdsum[b] / cnt, gdmax[b], logf(cnt) };

    float h[DIM];
    for (int d = 0; d < DIM; ++d) h[d] = rb1[d];
    for (int g = 0; g < DIM; ++g) {
        float v = gsum[b * DIM + g] / cnt;                 // mean pool
        for (int d = 0; d < DIM; ++d) h[d] += v * rw1[g * DIM + d];
    }
    for (int g = 0; g < DIM; ++g) {
        float v = gmax[b * DIM + g];                       // max pool
        for (int d = 0; d < DIM; ++d) h[d] += v * rw1[(DIM + g) * DIM + d];
    }
    for (int g = 0; g < 4; ++g) {
        float v = stats[g];
        for (int d = 0; d < DIM; ++d) h[d] += v * rw1[(2 * DIM + g) * DIM + d];
    }
    for (int d = 0; d < DIM; ++d) h[d] = fmaxf(h[d], 0.0f);

    float h2[DIM / 2];
    for (int d = 0; d < DIM / 2; ++d) {
        float a = rb2[d];
        for (int g = 0; g < DIM; ++g) a += h[g] * rw2[g * (DIM / 2) + d];
        h2[d] = fmaxf(a, 0.0f);
    }
    float t = temp[0];
    float s[NEXP]; float mx = -INFINITY;
    for (int e = 0; e < NEXP; ++e) {
        float a = rb3[e];
        for (int g = 0; g < DIM / 2; ++g) a += h2[g] * rw3[g * NEXP + e];
        s[e] = a / t; mx = fmaxf(mx, s[e]);
    }
    float den = 0.0f, p[NEXP];
    for (int e = 0; e < NEXP; ++e) { p[e] = expf(s[e] - mx); den += p[e]; }
    for (int e = 0; e < NEXP; ++e) gates[b * NEXP + e] = p[e] / den;
}

// ------------------------------------------ edge gather-scatter (GCN agg)
// one 32-lane wave per edge: lane handles 4 contiguous features
__global__ void k_edge_agg(const float* __restrict__ x,
                           const int* __restrict__ src, const int* __restrict__ dst,
                           const float* __restrict__ deg, long long E,
                           float* __restrict__ agg) {
    long long t = (long long)blockIdx.x * blockDim.x + threadIdx.x;
    long long e = t >> 5;
    int lane = (int)(t & 31);
    if (e >= E) return;
    int s = src[e], d = dst[e];
    float coef = rsqrtf(deg[s] + 1.0f) * rsqrtf(deg[d] + 1.0f);
    const float4* xs = (const float4*)(x + (long long)s * DIM);
    float4 v = xs[lane];
    float* ad = agg + (long long)d * DIM + lane * 4;
    atomicAdd(ad + 0, v.x * coef);
    atomicAdd(ad + 1, v.y * coef);
    atomicAdd(ad + 2, v.z * coef);
    atomicAdd(ad + 3, v.w * coef);
}

// ---------------------------------- experts GEMM + gate-combine (WMMA f32)
// one wave per 16-row tile; self-loop term folded into A load.
// wexp_p: expert weights repacked as K=512 (4 experts stacked), N=128.
__global__ void __launch_bounds__(32)
k_expert(const float* __restrict__ x, const float* __restrict__ agg,
         const float* __restrict__ deg, const int* __restrict__ batch,
         const float* __restrict__ gates,
         const v2f* __restrict__ wexp_p, const float* __restrict__ expert_b,
         int N, float* __restrict__ h1_out, float* colsum, float* colsq) {
    int lane = threadIdx.x, half = lane >> 4, mrow = lane & 15;
    int r0 = blockIdx.x * 16;
    __builtin_prefetch(wexp_p, 0, 1);

    // A fragments: 16x128 tile of (agg + x/deg), f32 WMMA lane layout
    v2f afrag[32];
    {
        int row = r0 + mrow; if (row >= N) row = N - 1;
        float invd = 1.0f / (deg[row] + 1.0f);
        const v2f* av = (const v2f*)(agg + (long long)row * DIM);
        const v2f* xv = (const v2f*)(x + (long long)row * DIM);
#pragma unroll
        for (int s2 = 0; s2 < 32; ++s2) {
            v2f a = av[2 * s2 + half], xx = xv[2 * s2 + half];
            afrag[s2].x = a.x + xx.x * invd;
            afrag[s2].y = a.y + xx.y * invd;
        }
    }
    // per-row gates (rows of C layout: M = i + 8*half)
    float gt[8][NEXP]; bool valid[8];
#pragma unroll
    for (int i = 0; i < 8; ++i) {
        int row = r0 + i + 8 * half;
        valid[i] = row < N;
        int b = batch[valid[i] ? row : (N - 1)];
#pragma unroll
        for (int e = 0; e < NEXP; ++e) gt[i][e] = gates[b * NEXP + e];
    }

    v8f vz = {};
    for (int j = 0; j < 8; ++j) {                 // output column tiles
        v8f acc[NEXP];
#pragma unroll
        for (int e = 0; e < NEXP; ++e) acc[e] = vz;
        int col = j * 16 + mrow;
        // pair index base: ((j*128 + e*32 + s2)*2 + half)*16 + mrow
#pragma unroll 4
        for (int s2 = 0; s2 < 32; ++s2) {
#pragma unroll
            for (int e = 0; e < NEXP; ++e) {
                v2f bf = wexp_p[((j * 128 + e * 32 + s2) * 2 + half) * 16 + mrow];
                acc[e] = __builtin_amdgcn_wmma_f32_16x16x4_f32(
                    false, afrag[s2], false, bf, (short)0, acc[e], false, false);
            }
        }
        float bias[NEXP];
#pragma unroll
        for (int e = 0; e < NEXP; ++e) bias[e] = expert_b[e * DIM + col];
        float csum = 0.0f, csq = 0.0f;
#pragma unroll
        for (int i = 0; i < 8; ++i) {
            int row = r0 + i + 8 * half;
            float kv = 0.0f;
#pragma unroll
            for (int e = 0; e < NEXP; ++e) kv += gt[i][e] * (acc[e][i] + bias[e]);
            if (valid[i]) {
                float h = x[(long long)row * DIM + col] + kv;   // residual
                h1_out[(long long)row * DIM + col] = h;
                csum += h; csq += h * h;
            }
        }
        atomicAdd(&colsum[col], csum);
        atomicAdd(&colsq[col], csq);
    }
}

// --------------------------------------------------- batchnorm finalizer
__global__ void k_bn_finalize(const float* colsum, const float* colsq,
                              const float* __restrict__ g, const float* __restrict__ b,
                              float invN, float* scale, float* shift) {
    int d = threadIdx.x;
    if (d >= DIM) return;
    float mu = colsum[d] * invN;
    float var = colsq[d] * invN - mu * mu;
    float r = rsqrtf(var + EPSV);
    float sc = g[d] * r;
    scale[d] = sc;
    shift[d] = b[d] - mu * sc;
}

// --------------------------------------- FFN: bn1 -> GEMM -> relu -> GEMM
__global__ void __launch_bounds__(32)
k_ffn(const float* __restrict__ h1, const float* __restrict__ bn1s,
      const float* __restrict__ bn1t,
      const v2f* __restrict__ w1p, const float* __restrict__ b1,
      const v2f* __restrict__ w2p, const float* __restrict__ b2,
      int N, float* __restrict__ h2_out, float* colsum, float* colsq) {
    __shared__ float lds[16 * 16];                // D-layout -> A-layout staging
    int lane = threadIdx.x, half = lane >> 4, mrow = lane & 15;
    int r0 = blockIdx.x * 16;
    __builtin_prefetch(w1p, 0, 1);

    // A fragments of bn1(h1)
    v2f afrag[32];
    {
        int row = r0 + mrow; if (row >= N) row = N - 1;
        const v2f* hv = (const v2f*)(h1 + (long long)row * DIM);
        const v2f* sv = (const v2f*)bn1s;
        const v2f* tv = (const v2f*)bn1t;
#pragma unroll
        for (int s2 = 0; s2 < 32; ++s2) {
            v2f h = hv[2 * s2 + half], sc = sv[2 * s2 + half], sh = tv[2 * s2 + half];
            afrag[s2].x = h.x * sc.x + sh.x;
            afrag[s2].y = h.y * sc.y + sh.y;
        }
    }
    v8f vz = {};
    v8f acc2[8];
#pragma unroll
    for (int jo = 0; jo < 8; ++jo) acc2[jo] = vz;

    for (int j = 0; j < 16; ++j) {                 // 256 hidden cols = K of GEMM2
        v8f acc1 = vz;
        int col1 = j * 16 + mrow;
        // w1 pairs: ((j*32 + s2)*2 + half)*16 + mrow   (K=128, 16 col tiles)
#pragma unroll 4
        for (int s2 = 0; s2 < 32; ++s2) {
            v2f bf = w1p[((j * 32 + s2) * 2 + half) * 16 + mrow];
            acc1 = __builtin_amdgcn_wmma_f32_16x16x4_f32(
                false, afrag[s2], false, bf, (short)0, acc1, false, false);
        }
        float bb = b1[col1];
        __syncthreads();                           // prior-iter LDS reads done
#pragma unroll
        for (int i = 0; i < 8; ++i) {
            float v = fmaxf(acc1[i] + bb, 0.0f);   // relu
            lds[(i + 8 * half) * 16 + mrow] = v;   // [row][k-in-tile]
        }
        __syncthreads();
        v2f a2[4];
#pragma unroll
        for (int t = 0; t < 4; ++t) {
            a2[t].x = lds[mrow * 16 + 4 * t + 2 * half];
            a2[t].y = lds[mrow * 16 + 4 * t + 2 * half + 1];
        }
        // w2 pairs: ((jo*64 + j*4 + t)*2 + half)*16 + mrow  (K=256, 8 col tiles)
#pragma unroll
        for (int t = 0; t < 4; ++t) {
#pragma unroll
            for (int jo = 0; jo < 8; ++jo) {
                v2f bf = w2p[((jo * 64 + j * 4 + t) * 2 + half) * 16 + mrow];
                acc2[jo] = __builtin_amdgcn_wmma_f32_16x16x4_f32(
                    false, a2[t], false, bf, (short)0, acc2[jo], false, false);
            }
        }
    }
    // epilogue: residual + column stats for bn2
    for (int jo = 0; jo < 8; ++jo) {
        int col = jo * 16 + mrow;
        float bb = b2[col], sc = bn1s[col], sh = bn1t[col];
        float csum = 0.0f, csq = 0.0f;
#pragma unroll
        for (int i = 0; i < 8; ++i) {
            int row = r0 + i + 8 * half;
            if (row < N) {
                float hbn = h1[(long long)row * DIM + col] * sc + sh;
                float h2 = hbn + acc2[jo][i] + bb;
                h2_out[(long long)row * DIM + col] = h2;
                csum += h2; csq += h2 * h2;
            }
        }
        atomicAdd(&colsum[col], csum);
        atomicAdd(&colsq[col], csq);
    }
}

// --------------------------------------------------------- final bn apply
__global__ void k_apply_bn(const float* __restrict__ h2, const float* __restrict__ scale,
                           const float* __restrict__ shift, long long total,
                           float* __restrict__ out) {
    long long i = (long long)blockIdx.x * blockDim.x + threadIdx.x;
    if (i >= total) return;
    int d = (int)(i % DIM);
    out[i] = h2[i] * scale[d] + shift[d];
}

// ------------------------------------------------------------------ driver
extern "C" void kernel_launch(void* const* d_in, const int* in_sizes, int n_in,
                              void* d_out, int out_size, void* d_ws, size_t ws_size,
                              hipStream_t stream) {
    const float* x           = (const float*)d_in[0];
    const int*   edge        = (const int*)  d_in[1];
    const int*   batch       = (const int*)  d_in[2];
    const float* temperature = (const float*)d_in[3];
    const float* rw1 = (const float*)d_in[4];
    const float* rb1 = (const float*)d_in[5];
    const float* rw2 = (const float*)d_in[6];
    const float* rb2 = (const float*)d_in[7];
    const float* rw3 = (const float*)d_in[8];
    const float* rb3 = (const float*)d_in[9];
    const float* expert_w = (const float*)d_in[10];
    const float* expert_b = (const float*)d_in[11];
    const float* bn1_g = (const float*)d_in[12];
    const float* bn1_b = (const float*)d_in[13];
    const float* ff_w1 = (const float*)d_in[14];
    const float* ff_b1 = (const float*)d_in[15];
    const float* ff_w2 = (const float*)d_in[16];
    const float* ff_b2 = (const float*)d_in[17];
    const float* bn2_g = (const float*)d_in[18];
    const float* bn2_b = (const float*)d_in[19];

    const int N = in_sizes[0] / DIM;
    const long long E = in_sizes[1] / 2;
    const int* src = edge;
    const int* dst = edge + E;

    float* ws = (float*)d_ws;
    size_t off = 0;
    float* deg     = ws + off; off += (size_t)N;
    float* gsum    = ws + off; off += (size_t)NGRAPH * DIM;
    float* gmax    = ws + off; off += (size_t)NGRAPH * DIM;
    float* gcnt    = ws + off; off += NGRAPH;
    float* gdsum   = ws + off; off += NGRAPH;
    float* gdmax   = ws + off; off += NGRAPH;
    float* gates   = ws + off; off += NGRAPH * NEXP;
    float* colsum1 = ws + off; off += DIM;
    float* colsq1  = ws + off; off += DIM;
    float* colsum2 = ws + off; off += DIM;
    float* colsq2  = ws + off; off += DIM;
    float* bn1s    = ws + off; off += DIM;
    float* bn1t    = ws + off; off += DIM;
    float* bn2s    = ws + off; off += DIM;
    float* bn2t    = ws + off; off += DIM;
    const size_t small_total = off;            // everything above starts zeroed
    v2f* wexp_p = (v2f*)(ws + off); off += (size_t)NEXP * DIM * DIM;  // 512x128 pairs
    v2f* w1p    = (v2f*)(ws + off); off += (size_t)DIM * 2 * DIM;     // 128x256 pairs
    v2f* w2p    = (v2f*)(ws + off); off += (size_t)2 * DIM * DIM;     // 256x128 pairs
    float* agg = ws + off; off += (size_t)N * DIM;   // reused as h2 buffer
    float* h1 = (float*)d_out;                 // h1 lives in d_out until final BN2
    float* h2 = agg;

    const long long ND = (long long)N * DIM;
    const int TPB = 256;

    // init accumulation buffers (every call -> deterministic replays)
    k_fill<<<(int)((small_total + TPB - 1) / TPB), TPB, 0, stream>>>(ws, (long long)small_total, 0.0f);
    k_fill<<<(NGRAPH * DIM + TPB - 1) / TPB, TPB, 0, stream>>>(gmax, NGRAPH * DIM, -INFINITY);
    k_fill<<<(NGRAPH + TPB - 1) / TPB, TPB, 0, stream>>>(gdmax, NGRAPH, -INFINITY);
    k_fill<<<(int)((ND + TPB - 1) / TPB), TPB, 0, stream>>>(agg, ND, 0.0f);

    // repack weights into coalesced WMMA B-fragment pair layout
    k_repack<<<(512 * 128 / 2 + TPB - 1) / TPB, TPB, 0, stream>>>(expert_w, 512, 128, wexp_p);
    k_repack<<<(128 * 256 / 2 + TPB - 1) / TPB, TPB, 0, stream>>>(ff_w1, 128, 256, w1p);
    k_repack<<<(256 * 128 / 2 + TPB - 1) / TPB, TPB, 0, stream>>>(ff_w2, 256, 128, w2p);

    k_deg<<<(int)((E + TPB - 1) / TPB), TPB, 0, stream>>>(dst, (int)E, deg);
    k_node_stats<<<(int)((ND + TPB - 1) / TPB), TPB, 0, stream>>>(
        x, batch, deg, N, gsum, gmax, gcnt, gdsum, gdmax);
    k_router<<<1, NGRAPH, 0, stream>>>(gsum, gmax, gcnt, gdsum, gdmax, temperature,
                                       rw1, rb1, rw2, rb2, rw3, rb3, gates);
    k_edge_agg<<<(int)((E * 32 + TPB - 1) / TPB), TPB, 0, stream>>>(
        x, src, dst, deg, E, agg);

    const int ntiles = (N + 15) / 16;
    k_expert<<<ntiles, 32, 0, stream>>>(x, agg, deg, batch, gates,
                                        wexp_p, expert_b, N, h1, colsum1, colsq1);
    k_bn_finalize<<<1, DIM, 0, stream>>>(colsum1, colsq1, bn1_g, bn1_b, 1.0f / N, bn1s, bn1t);
    k_ffn<<<ntiles, 32, 0, stream>>>(h1, bn1s, bn1t, w1p, ff_b1, w2p, ff_b2,
                                     N, h2, colsum2, colsq2);
    k_bn_finalize<<<1, DIM, 0, stream>>>(colsum2, colsq2, bn2_g, bn2_b, 1.0f / N, bn2s, bn2t);
    k_apply_bn<<<(int)((ND + TPB - 1) / TPB), TPB, 0, stream>>>(h2, bn2s, bn2t, ND, (float*)d_out);
}